// Conv1dLocal_51969104281954
// MI455X (gfx1250) — compile-verified
//
#include <hip/hip_runtime.h>

typedef __attribute__((ext_vector_type(2))) float v2f;
typedef __attribute__((ext_vector_type(8))) float v8f;
typedef __attribute__((ext_vector_type(4))) unsigned int u32x4;
typedef __attribute__((ext_vector_type(8))) unsigned int u32x8;

namespace {
constexpr int kB       = 32;
constexpr int kInCh    = 64;
constexpr int kOutCh   = 64;
constexpr int kOutSize = 512;
constexpr int kKW      = 7;
constexpr int kL       = (kOutSize - 1) + kKW;   // 518
constexpr int kIcChunk = 16;                     // input channels per LDS pass
constexpr int kKch     = kIcChunk * kKW;         // 112 = K per chunk (TDM packs rows densely)
constexpr int kChunks  = kInCh / kIcChunk;       // 4
constexpr int kThreads = 256;                    // 8 wave32s per block
}

// Issue one 3D TDM load:  tile = 7 (k, contiguous) x tile1 (inner, stride0) x tile2 (outer, stride1)
// All inputs are uniform (blockIdx-derived) -> SGPR descriptor groups.
__device__ __forceinline__ void tdm_load_3d(unsigned lds_addr, const float* gptr,
                                            unsigned tile1, unsigned long long stride0_elems,
                                            unsigned tile2, unsigned long long stride1_elems) {
    const unsigned long long ga = (unsigned long long)gptr;

    u32x4 g0 = {0u, 0u, 0u, 0u};
    g0[0] = 1u;                                             // count=1, user descriptor
    g0[1] = lds_addr;                                       // LDS byte address
    g0[2] = (unsigned)(ga & 0xFFFFFFFFull);                 // global_addr[31:0]
    g0[3] = (unsigned)((ga >> 32) & 0x01FFFFFFull)          // global_addr[56:32]
          | (2u << 30);                                     // type=2 ("image")

    const unsigned d0 = kKW;                                // tensor_dim0 = tile_dim0 = 7
    u32x8 g1 = {0u, 0u, 0u, 0u, 0u, 0u, 0u, 0u};
    g1[0] = 2u << 16;                                       // data_size=4B; no multicast/pad/iterate
    g1[1] = (d0 & 0xFFFFu) << 16;                           // tensor_dim0[15:0]
    g1[2] = (d0 >> 16) | ((tile1 & 0xFFFFu) << 16);         // tensor_dim0[31:16] | tensor_dim1[15:0]
    g1[3] = (tile1 >> 16) | ((unsigned)kKW << 16);          // tensor_dim1[31:16] | tile_dim0
    g1[4] = (tile1 & 0xFFFFu) | ((tile2 & 0xFFFFu) << 16);  // tile_dim1 | tile_dim2
    g1[5] = (unsigned)(stride0_elems & 0xFFFFFFFFull);      // tensor_dim0_stride[31:0]
    g1[6] = (unsigned)((stride0_elems >> 32) & 0xFFFFull)   // tensor_dim0_stride[47:32]
          | ((unsigned)(stride1_elems & 0xFFFFull) << 16);  // tensor_dim1_stride[15:0]
    g1[7] = (unsigned)((stride1_elems >> 16) & 0xFFFFFFFFull); // tensor_dim1_stride[47:16]

    u32x4 g2 = {0u, 0u, 0u, 0u};
    g2[0] = tile2;                                          // tensor_dim2
    u32x4 g3 = {0u, 0u, 0u, 0u};                            // tile_dim3/4 = 0 (unused)

    asm volatile("tensor_load_to_lds %0, %1, %2, %3"
                 :: "s"(g0), "s"(g1), "s"(g2), "s"(g3)
                 : "memory");
}

__device__ __forceinline__ unsigned lds_byte_addr(const void* p) {
    // Generic (flat) address of a __shared__ object: low 32 bits are the LDS byte offset.
    return (unsigned)(unsigned long long)p;
}

__global__ __launch_bounds__(kThreads)
void conv1d_local_wmma(const float* __restrict__ x,
                       const float* __restrict__ w,
                       float* __restrict__ out) {
    // Double-buffered TDM landing zones (dense rows: stride = 112 f32). 84 KB of 320 KB WGP LDS.
    __shared__ float Xlds[2][kB * kKch];      // 2 x 14336 B  (A tile, M x K)
    __shared__ float Wlds[2][kOutCh * kKch];  // 2 x 28672 B  (B^T tile, N x K)

    const int s    = blockIdx.x;        // output position 0..511
    const int tid  = threadIdx.x;
    const int lane = tid & 31;          // wave32
    const int wave = tid >> 5;          // 0..7
    const int mt   = wave >> 2;         // batch tile   0..1
    const int nt   = wave & 3;          // outch tile   0..3
    const int half = lane >> 4;
    const int l16  = lane & 15;

    // Global tile bases for chunk ic0:
    //   X: x + ic0*kL + s                 rows: i (stride kL=518),    planes: b (stride 64*518)
    //   W: w + ic0*3584 + s*7             rows: i (stride 3584),      planes: o (stride 229376)
    const float* xbase = x + s;
    const float* wbase = w + (size_t)s * kKW;

    // ---- prologue: kick off chunk 0 ----
    if (wave == 0) {
        tdm_load_3d(lds_byte_addr(&Xlds[0][0]), xbase + 0 * kIcChunk * kL,
                    kIcChunk, (unsigned long long)kL, kB,
                    (unsigned long long)kInCh * kL);
    } else if (wave == 1) {
        tdm_load_3d(lds_byte_addr(&Wlds[0][0]), wbase + 0 * kIcChunk * (kOutSize * kKW),
                    kIcChunk, (unsigned long long)(kOutSize * kKW), kOutCh,
                    (unsigned long long)kInCh * kOutSize * kKW);
    }

    v8f acc = {};

#pragma unroll
    for (int c = 0; c < kChunks; ++c) {
        const int cur = c & 1;
        const int nxt = cur ^ 1;

        // ---- prefetch chunk c+1 into the other buffer (overlaps with WMMA on chunk c) ----
        if (c + 1 < kChunks) {
            if (wave == 0) {
                tdm_load_3d(lds_byte_addr(&Xlds[nxt][0]),
                            xbase + (c + 1) * kIcChunk * kL,
                            kIcChunk, (unsigned long long)kL, kB,
                            (unsigned long long)kInCh * kL);
            } else if (wave == 1) {
                tdm_load_3d(lds_byte_addr(&Wlds[nxt][0]),
                            wbase + (size_t)(c + 1) * kIcChunk * (kOutSize * kKW),
                            kIcChunk, (unsigned long long)(kOutSize * kKW), kOutCh,
                            (unsigned long long)kInCh * kOutSize * kKW);
            }
            // TDM completes in-order per wave: <=1 outstanding means chunk c has landed.
            __builtin_amdgcn_s_wait_tensorcnt(1);
        } else {
            __builtin_amdgcn_s_wait_tensorcnt(0);
        }
        __syncthreads();   // publish buf[cur] to all waves

        // A 16x4 f32: lane(<16) holds (M=l16, K=k0,k0+1); lane(>=16): K=k0+2,k0+3.
        const float* arow = &Xlds[cur][(mt * 16 + l16) * kKch + 2 * half];
        // B 4x16 f32 mirrors A: per lane two consecutive K at column N=l16.
        const float* brow = &Wlds[cur][(nt * 16 + l16) * kKch + 2 * half];

#pragma unroll
        for (int kk = 0; kk < kKch; kk += 4) {      // 28 WMMA steps per chunk
            v2f a = *(const v2f*)(arow + kk);       // ds_load_b64
            v2f b = *(const v2f*)(brow + kk);       // ds_load_b64
            acc = __builtin_amdgcn_wmma_f32_16x16x4_f32(
                false, a, false, b, (short)0, acc, false, false);
        }
        __syncthreads();   // all waves done reading buf[cur] before chunk c+2 overwrites it
    }

    // D 16x16 f32 layout: VGPR r -> (M=r, N=lane) lanes 0-15; (M=r+8, N=lane-16).
    const int n = nt * 16 + l16;
#pragma unroll
    for (int r = 0; r < 8; ++r) {
        const int b = mt * 16 + half * 8 + r;
        out[(size_t)b * (kOutCh * kOutSize) + (size_t)n * kOutSize + s] = acc[r];
    }
}

extern "C" void kernel_launch(void* const* d_in, const int* in_sizes, int n_in,
                              void* d_out, int out_size, void* d_ws, size_t ws_size,
                              hipStream_t stream) {
    (void)in_sizes; (void)n_in; (void)d_ws; (void)ws_size; (void)out_size;
    const float* x = (const float*)d_in[0];   // (32, 64, 518) f32
    const float* w = (const float*)d_in[1];   // (64, 64, 512, 7) f32
    float* out = (float*)d_out;               // (32, 64, 512) f32
    conv1d_local_wmma<<<dim3(kOutSize), dim3(kThreads), 0, stream>>>(x, w, out);
}